// SupplierGraphSAGE_47614007443729
// MI455X (gfx1250) — compile-verified
//
#include <hip/hip_runtime.h>
#include <hip/hip_bf16.h>
#include <math.h>

// ---------------------------------------------------------------------------
// SupplierGraphSAGE on MI455X (gfx1250, wave32).
//
// Pipeline (all L2-resident; 192MB L2 >> 77MB working set):
//   1. zero msg_sum / cnt / stats
//   2. edge scatter: atomicAdd x[src] -> msg_sum[dst], cnt[dst]++   (bandwidth limiter)
//   3. invcnt = 1/max(cnt,1)
//   4. SAGE GEMM via V_WMMA_F32_16X16X4_F32: tmp = (msg*inv)@W0l^T + b0l + x@W0r^T
//   5. BN stats (sum, sumsq) -> finalize (scale, shift) -> apply + ReLU + residual
//   6. repeat scatter/GEMM/BN for layer 1 (64 -> 32), write h to d_out
//   7. classifier 32->16->1 + sigmoid (LDS-staged weights) -> risk to d_out+N*32
// ---------------------------------------------------------------------------

#define N_NODES 100000
#define N_EDGES 1600000
#define BN_EPS  1e-5f

typedef __attribute__((ext_vector_type(2))) float v2f;
typedef __attribute__((ext_vector_type(8))) float v8f;

// ---------------------------------------------------------------------------
__global__ void zero_kernel(float* __restrict__ p, long long n) {
    long long i = (long long)blockIdx.x * blockDim.x + threadIdx.x;
    if (i < n) p[i] = 0.0f;
}

// ---------------------------------------------------------------------------
// One thread per (edge, channel). 64 consecutive threads cover one edge's 64
// channels -> coalesced reads of h[src], atomics into msg[dst] hit L2.
__global__ void scatter_kernel(const float* __restrict__ h,
                               const int* __restrict__ src,
                               const int* __restrict__ dst,
                               float* __restrict__ msg,
                               float* __restrict__ cnt,
                               int shift, int addCnt) {
    unsigned long long id = (unsigned long long)blockIdx.x * blockDim.x + threadIdx.x;
    int C = 1 << shift;
    unsigned e = (unsigned)(id >> shift);
    if (e >= N_EDGES) return;
    int c = (int)(id & (unsigned long long)(C - 1));
    int s = src[e];
    int d = dst[e];
    atomicAdd(&msg[(unsigned long long)d * C + c], h[(unsigned long long)s * C + c]);
    if (addCnt && c == 0) atomicAdd(&cnt[d], 1.0f);
}

__global__ void invcnt_kernel(float* __restrict__ cnt) {
    int i = blockIdx.x * blockDim.x + threadIdx.x;
    if (i < N_NODES) cnt[i] = 1.0f / fmaxf(cnt[i], 1.0f);
}

// ---------------------------------------------------------------------------
// SAGE linear layer on the matrix pipe.
// One wave -> 16-node x 16-outchannel f32 tile; accumulate K in steps of 4 with
// two chained V_WMMA_F32_16X16X4_F32 (neighbor-mean path + self path).
//
// A  (16x4 f32): lanes 0-15 row M=lane,   v0=K0 v1=K1 ; lanes 16-31 v0=K2 v1=K3
// B  (4x16 f32): lanes 0-15 col N=lane,   v0=K0 v1=K1 ; lanes 16-31 v0=K2 v1=K3
// C/D (16x16):   VGPR r, lanes 0-15 -> M=r, N=lane ; lanes 16-31 -> M=r+8
//
// blockDim.x = 32 * (OUTC/16);  gridDim.x = N/16  (N%16==0)
__global__ void sage_wmma_kernel(const float* __restrict__ msg,   // [N,K] neighbor sums
                                 const float* __restrict__ invc,  // [N]
                                 const float* __restrict__ hself, // [N,K]
                                 const float* __restrict__ Wl,    // [OUTC,K] row-major
                                 const float* __restrict__ Wr,    // [OUTC,K]
                                 const float* __restrict__ bias,  // [OUTC]
                                 float* __restrict__ out,         // [N,OUTC]
                                 int K, int OUTC) {
    int lane = threadIdx.x & 31;
    int wave = threadIdx.x >> 5;
    int nodeBase = blockIdx.x * 16;
    int outBase  = wave * 16;
    int col  = lane & 15;
    int half = lane >> 4;          // selects K pair {0,1} vs {2,3}
    int row  = nodeBase + col;     // A-row owned by this lane
    int n    = outBase + col;      // B-column owned by this lane
    float inv = invc[row];

    v8f acc = {0.f, 0.f, 0.f, 0.f, 0.f, 0.f, 0.f, 0.f};

    const float* pm = msg   + (unsigned long long)row * K;
    const float* ps = hself + (unsigned long long)row * K;
    const float* pl = Wl + (unsigned long long)n * K;
    const float* pr = Wr + (unsigned long long)n * K;

    for (int k0 = 0; k0 < K; k0 += 4) {
        int kA = k0 + half * 2;    // even -> aligned float2 loads
        v2f aM = *(const v2f*)(pm + kA);
        aM.x *= inv; aM.y *= inv;                 // fold mean-divide into A load
        v2f aS = *(const v2f*)(ps + kA);
        v2f bL = *(const v2f*)(pl + kA);          // B[k][n] = W[n][k]
        v2f bR = *(const v2f*)(pr + kA);
        acc = __builtin_amdgcn_wmma_f32_16x16x4_f32(
            false, aM, false, bL, (short)0, acc, false, false);
        acc = __builtin_amdgcn_wmma_f32_16x16x4_f32(
            false, aS, false, bR, (short)0, acc, false, false);
    }

    float bn = bias[n];            // bias depends only on output channel
#pragma unroll
    for (int r = 0; r < 8; ++r) {
        int node = nodeBase + r + 8 * half;
        out[(unsigned long long)node * OUTC + n] = acc[r] + bn;
    }
}

// ---------------------------------------------------------------------------
// BN stats: grid stride is a multiple of C, so each thread owns one channel;
// thread-local accumulation then 2 atomics/thread.
__global__ void bn_stats_kernel(const float* __restrict__ v,
                                float* __restrict__ stats,   // [sum C][sumsq C]
                                int shift, long long nElem) {
    int C = 1 << shift;
    long long t = (long long)blockIdx.x * blockDim.x + threadIdx.x;
    long long stride = (long long)gridDim.x * blockDim.x;  // multiple of C
    float s = 0.f, ss = 0.f;
    for (long long i = t; i < nElem; i += stride) {
        float x = v[i];
        s += x; ss += x * x;
    }
    int c = (int)(t & (C - 1));
    atomicAdd(&stats[c], s);
    atomicAdd(&stats[C + c], ss);
}

__global__ void bn_finalize_kernel(float* __restrict__ stats,
                                   const float* __restrict__ gamma,
                                   const float* __restrict__ beta, int C) {
    int c = threadIdx.x;
    if (c < C) {
        float invN = 1.0f / (float)N_NODES;
        float mu  = stats[c] * invN;
        float var = stats[C + c] * invN - mu * mu;   // biased, as torch BN train
        float sc  = gamma[c] * rsqrtf(var + BN_EPS);
        stats[2 * C + c] = sc;
        stats[3 * C + c] = beta[c] - mu * sc;
    }
}

__global__ void bn_apply_kernel(const float* __restrict__ pre,
                                const float* __restrict__ stats,
                                const float* __restrict__ residual,  // may be null
                                float* __restrict__ out,
                                int shift, long long nElem) {
    long long i = (long long)blockIdx.x * blockDim.x + threadIdx.x;
    if (i >= nElem) return;
    int C = 1 << shift;
    int c = (int)(i & (C - 1));
    float v = fmaxf(pre[i] * stats[2 * C + c] + stats[3 * C + c], 0.0f);
    if (residual) v += residual[i];
    out[i] = v;
}

// ---------------------------------------------------------------------------
// Node classifier: Linear(32,16) -> ReLU -> Linear(16,1) -> Sigmoid.
// Weights staged in LDS once per block.
__global__ void clf_kernel(const float* __restrict__ h,     // [N,32]
                           const float* __restrict__ Wc1,   // [16,32]
                           const float* __restrict__ bc1,   // [16]
                           const float* __restrict__ Wc2,   // [1,16]
                           const float* __restrict__ bc2,   // [1]
                           float* __restrict__ risk) {
    __shared__ float sW1[16 * 32];
    __shared__ float sB1[16];
    __shared__ float sW2[16];
    for (int k = threadIdx.x; k < 16 * 32; k += blockDim.x) sW1[k] = Wc1[k];
    if (threadIdx.x < 16) { sB1[threadIdx.x] = bc1[threadIdx.x];
                            sW2[threadIdx.x] = Wc2[threadIdx.x]; }
    __syncthreads();

    int i = blockIdx.x * blockDim.x + threadIdx.x;
    if (i >= N_NODES) return;
    float hr[32];
#pragma unroll
    for (int c = 0; c < 32; ++c) hr[c] = h[(unsigned long long)i * 32 + c];
    float z = bc2[0];
#pragma unroll
    for (int j = 0; j < 16; ++j) {
        float a = sB1[j];
#pragma unroll
        for (int c = 0; c < 32; ++c) a += sW1[j * 32 + c] * hr[c];
        z += sW2[j] * fmaxf(a, 0.0f);
    }
    risk[i] = 1.0f / (1.0f + __expf(-z));
}

// ---------------------------------------------------------------------------
extern "C" void kernel_launch(void* const* d_in, const int* in_sizes, int n_in,
                              void* d_out, int out_size, void* d_ws, size_t ws_size,
                              hipStream_t stream) {
    (void)in_sizes; (void)n_in; (void)out_size; (void)ws_size;

    const float* x    = (const float*)d_in[0];
    const int*   edge = (const int*)d_in[1];     // [2][E]: src row then dst row
    const float* W0l  = (const float*)d_in[2];
    const float* b0l  = (const float*)d_in[3];
    const float* W0r  = (const float*)d_in[4];
    const float* g0   = (const float*)d_in[5];
    const float* be0  = (const float*)d_in[6];
    const float* W1l  = (const float*)d_in[7];
    const float* b1l  = (const float*)d_in[8];
    const float* W1r  = (const float*)d_in[9];
    const float* g1   = (const float*)d_in[10];
    const float* be1  = (const float*)d_in[11];
    const float* Wc1  = (const float*)d_in[12];
    const float* bc1  = (const float*)d_in[13];
    const float* Wc2  = (const float*)d_in[14];
    const float* bc2  = (const float*)d_in[15];

    const int* src = edge;
    const int* dst = edge + N_EDGES;

    // workspace carve-up (floats)
    float* ws    = (float*)d_ws;
    float* msg   = ws;                                   // N*64
    float* invc  = msg  + (size_t)N_NODES * 64;          // N
    float* tmp   = invc + (size_t)N_NODES;               // N*64 (layer1 reuses N*32)
    float* h0    = tmp  + (size_t)N_NODES * 64;          // N*64
    float* stats = h0   + (size_t)N_NODES * 64;          // 256

    float* outH  = (float*)d_out;                        // [N,32]
    float* risk  = outH + (size_t)N_NODES * 32;          // [N]

    const long long msgElems = (long long)N_NODES * 64;
    const int ZB = 256;

    // ---- layer 0 -----------------------------------------------------------
    zero_kernel<<<(unsigned)((msgElems + ZB - 1) / ZB), ZB, 0, stream>>>(msg, msgElems);
    zero_kernel<<<(N_NODES + ZB - 1) / ZB, ZB, 0, stream>>>(invc, N_NODES);
    zero_kernel<<<1, 256, 0, stream>>>(stats, 256);

    unsigned scatterGrid = (unsigned)(((long long)N_EDGES * 64 + 255) / 256);
    scatter_kernel<<<scatterGrid, 256, 0, stream>>>(x, src, dst, msg, invc, 6, 1);
    invcnt_kernel<<<(N_NODES + 255) / 256, 256, 0, stream>>>(invc);

    sage_wmma_kernel<<<N_NODES / 16, 128, 0, stream>>>(msg, invc, x, W0l, W0r, b0l,
                                                       tmp, 64, 64);

    bn_stats_kernel<<<512, 256, 0, stream>>>(tmp, stats, 6, msgElems);
    bn_finalize_kernel<<<1, 64, 0, stream>>>(stats, g0, be0, 64);
    bn_apply_kernel<<<(unsigned)((msgElems + 255) / 256), 256, 0, stream>>>(
        tmp, stats, x, h0, 6, msgElems);                 // h0 = x + relu(BN(...))

    // ---- layer 1 -----------------------------------------------------------
    zero_kernel<<<(unsigned)((msgElems + ZB - 1) / ZB), ZB, 0, stream>>>(msg, msgElems);
    zero_kernel<<<1, 256, 0, stream>>>(stats, 256);

    scatter_kernel<<<scatterGrid, 256, 0, stream>>>(h0, src, dst, msg, invc, 6, 0);

    sage_wmma_kernel<<<N_NODES / 16, 64, 0, stream>>>(msg, invc, h0, W1l, W1r, b1l,
                                                      tmp, 64, 32);

    const long long l1Elems = (long long)N_NODES * 32;
    bn_stats_kernel<<<512, 256, 0, stream>>>(tmp, stats, 5, l1Elems);
    bn_finalize_kernel<<<1, 32, 0, stream>>>(stats, g1, be1, 32);
    bn_apply_kernel<<<(unsigned)((l1Elems + 255) / 256), 256, 0, stream>>>(
        tmp, stats, nullptr, outH, 5, l1Elems);          // h -> d_out

    // ---- classifier --------------------------------------------------------
    clf_kernel<<<(N_NODES + 255) / 256, 256, 0, stream>>>(outH, Wc1, bc1, Wc2, bc2, risk);
}